// Ultimus_Block_28157805592774
// MI455X (gfx1250) — compile-verified
//
#include <hip/hip_runtime.h>
#include <math.h>

#define N_ROWS 16384
#define DIM 48
#define HEADS 8
#define TILE 16
#define NTILES (N_ROWS / TILE)            // 1024
#define SPLITS 8
#define TILES_PER_SPLIT (NTILES / SPLITS) // 128
#define WAVES_PER_BLOCK 8
#define SQRT_H 2.8284271247461903f

typedef __attribute__((ext_vector_type(2))) float v2f;
typedef __attribute__((ext_vector_type(8))) float v8f;

// ---------------------------------------------------------------------------
// Kernel 1: QKV projection.  q/k/v[row][h] = b[h] + sum_d x[row][d]*W[h][d]
// ---------------------------------------------------------------------------
__global__ __launch_bounds__(256) void qkv_kernel(
    const float* __restrict__ x,
    const float* __restrict__ Wk, const float* __restrict__ bk,
    const float* __restrict__ Wq, const float* __restrict__ bq,
    const float* __restrict__ Wv, const float* __restrict__ bv,
    float* __restrict__ Q, float* __restrict__ K, float* __restrict__ V)
{
    const int row = blockIdx.x * blockDim.x + threadIdx.x;   // grid == N_ROWS
    const float* xp = x + (size_t)row * DIM;
    float xr[DIM];
#pragma unroll
    for (int d = 0; d < DIM; d += 4) {
        float4 t = *(const float4*)(xp + d);
        xr[d+0] = t.x; xr[d+1] = t.y; xr[d+2] = t.z; xr[d+3] = t.w;
    }
#pragma unroll
    for (int h = 0; h < HEADS; ++h) {
        float aq = bq[h], ak = bk[h], av = bv[h];
#pragma unroll
        for (int d = 0; d < DIM; ++d) {
            const float xv = xr[d];
            aq = fmaf(xv, Wq[h*DIM + d], aq);
            ak = fmaf(xv, Wk[h*DIM + d], ak);
            av = fmaf(xv, Wv[h*DIM + d], av);
        }
        Q[(size_t)row*HEADS + h] = aq;
        K[(size_t)row*HEADS + h] = ak;
        V[(size_t)row*HEADS + h] = av;
    }
}

// ---------------------------------------------------------------------------
// Kernel 2: flash-attention partial pass.
// One wave per (16-row Q tile, j-split). Computes S^T = K_tile @ Q_tile^T
// with two V_WMMA_F32_16X16X4_F32 (K=8 -> 2 chunks of 4).
// WMMA f32 A/B operand layout: (m|n, k): vgpr = k&1, lane-half = (k>=2).
// The two f32 values a lane contributes per K-chunk are CONTIGUOUS in the
// row (offsets 2*half and 4+2*half), so we fold `2*half` into the pointer
// and load them as b64 -- no register-array indexing / cndmask chains.
// Result layout: lane l holds query column i = i0 + (l&15); rows
// j_local = p + 8*(l>>4) in accumulator VGPR p. Each lane runs an online
// softmax over its 8 j's per tile and accumulates z[h] += e*V[j][h] on VALU.
// Lane halves merged once at the end via shfl_xor(16).
// ---------------------------------------------------------------------------
__global__ __launch_bounds__(256) void flash_kernel(
    const float* __restrict__ Q, const float* __restrict__ K,
    const float* __restrict__ V, float* __restrict__ part)
{
    const int lane  = threadIdx.x & 31;
    const int wave  = threadIdx.x >> 5;
    const int r     = lane & 15;
    const int half  = lane >> 4;
    const int itile = blockIdx.x * WAVES_PER_BLOCK + wave;
    const int i0    = itile * TILE;
    const int split = blockIdx.y;
    const int jt0   = split * TILES_PER_SPLIT;

    // B operand (fixed): Q^T tile; lane supplies column n = r, k = vgpr+2*half.
    const float* qp = Q + (size_t)(i0 + r) * HEADS + 2*half;
    const v2f b0 = *(const v2f*)(qp);       // Q[i0+r][2h], Q[i0+r][2h+1]
    const v2f b1 = *(const v2f*)(qp + 4);   // Q[i0+r][4+2h], Q[i0+r][5+2h]

    float m = -INFINITY;
    float l = 0.0f;
    float z[HEADS];
#pragma unroll
    for (int h = 0; h < HEADS; ++h) z[h] = 0.0f;

    // per-lane bases, hoisted out of the loop
    const float* kp = K + (size_t)(jt0*TILE + r) * HEADS + 2*half;
    const float* vp = V + (size_t)(jt0*TILE + 8*half) * HEADS;

    for (int jt = 0; jt < TILES_PER_SPLIT; ++jt) {
        // A operand: K tile rows; lane supplies row m = r, k = vgpr+2*half.
        const v2f a0 = *(const v2f*)(kp);
        const v2f a1 = *(const v2f*)(kp + 4);

        // prefetch next tile's K/V stream
        __builtin_prefetch(kp + TILE * HEADS, 0, 3);
        __builtin_prefetch(vp + TILE * HEADS, 0, 3);

        const v8f c0 = {0.f, 0.f, 0.f, 0.f, 0.f, 0.f, 0.f, 0.f};
        v8f st = __builtin_amdgcn_wmma_f32_16x16x4_f32(
            false, a0, false, b0, (short)0, c0, false, false);
        st = __builtin_amdgcn_wmma_f32_16x16x4_f32(
            false, a1, false, b1, (short)0, st, false, false);

        // online softmax over st[0..7] (this lane's 8 j values)
        float t01 = fmaxf(st[0], st[1]);
        float t23 = fmaxf(st[2], st[3]);
        float t45 = fmaxf(st[4], st[5]);
        float t67 = fmaxf(st[6], st[7]);
        float tmax = fmaxf(fmaxf(t01, t23), fmaxf(t45, t67));
        float mn   = fmaxf(m, tmax);
        float corr = __expf(m - mn);
        l *= corr;
#pragma unroll
        for (int h = 0; h < HEADS; ++h) z[h] *= corr;
#pragma unroll
        for (int p = 0; p < 8; ++p) {
            float e = __expf(st[p] - mn);
            l += e;
            float4 v0 = *(const float4*)(vp + p*HEADS);
            float4 v1 = *(const float4*)(vp + p*HEADS + 4);
            z[0] = fmaf(e, v0.x, z[0]);
            z[1] = fmaf(e, v0.y, z[1]);
            z[2] = fmaf(e, v0.z, z[2]);
            z[3] = fmaf(e, v0.w, z[3]);
            z[4] = fmaf(e, v1.x, z[4]);
            z[5] = fmaf(e, v1.y, z[5]);
            z[6] = fmaf(e, v1.z, z[6]);
            z[7] = fmaf(e, v1.w, z[7]);
        }
        m = mn;
        kp += TILE * HEADS;
        vp += TILE * HEADS;
    }

    // merge the two lane halves (j 0..7 with j 8..15 of each tile)
    float m2 = __shfl_xor(m, 16, 32);
    float M  = fmaxf(m, m2);
    float ca = __expf(m  - M);
    float cb = __expf(m2 - M);
    float L  = l * ca + __shfl_xor(l, 16, 32) * cb;
    float zt[HEADS];
#pragma unroll
    for (int h = 0; h < HEADS; ++h)
        zt[h] = z[h] * ca + __shfl_xor(z[h], 16, 32) * cb;

    if (half == 0) {
        float* pp = part + ((size_t)split * N_ROWS + (i0 + r)) * 10;
        pp[0] = M;
        pp[1] = L;
#pragma unroll
        for (int h = 0; h < HEADS; ++h) pp[2 + h] = zt[h];
    }
}

// ---------------------------------------------------------------------------
// Kernel 3: combine split partials, normalize by (L * sqrt(8)), apply the
// output projection Wo [D,H] + bo and the residual.
// ---------------------------------------------------------------------------
__global__ __launch_bounds__(256) void combine_kernel(
    const float* __restrict__ x, const float* __restrict__ Wo,
    const float* __restrict__ bo, const float* __restrict__ part,
    float* __restrict__ out)
{
    const int row = blockIdx.x * blockDim.x + threadIdx.x;  // grid == N_ROWS
    float M = -INFINITY;
#pragma unroll
    for (int s = 0; s < SPLITS; ++s)
        M = fmaxf(M, part[((size_t)s * N_ROWS + row) * 10]);
    float L = 0.0f;
    float z[HEADS];
#pragma unroll
    for (int h = 0; h < HEADS; ++h) z[h] = 0.0f;
#pragma unroll
    for (int s = 0; s < SPLITS; ++s) {
        const float* pp = part + ((size_t)s * N_ROWS + row) * 10;
        float c = __expf(pp[0] - M);
        L = fmaf(pp[1], c, L);
#pragma unroll
        for (int h = 0; h < HEADS; ++h) z[h] = fmaf(pp[2 + h], c, z[h]);
    }
    const float inv = 1.0f / (L * SQRT_H);
#pragma unroll
    for (int h = 0; h < HEADS; ++h) z[h] *= inv;

    const float* xp = x + (size_t)row * DIM;
    float* op = out + (size_t)row * DIM;
#pragma unroll
    for (int d = 0; d < DIM; ++d) {
        float acc = xp[d] + bo[d];
#pragma unroll
        for (int h = 0; h < HEADS; ++h)
            acc = fmaf(z[h], Wo[d*HEADS + h], acc);
        op[d] = acc;
    }
}

// ---------------------------------------------------------------------------
extern "C" void kernel_launch(void* const* d_in, const int* in_sizes, int n_in,
                              void* d_out, int out_size, void* d_ws, size_t ws_size,
                              hipStream_t stream)
{
    const float* x  = (const float*)d_in[0];
    const float* Wk = (const float*)d_in[1];
    const float* bk = (const float*)d_in[2];
    const float* Wq = (const float*)d_in[3];
    const float* bq = (const float*)d_in[4];
    const float* Wv = (const float*)d_in[5];
    const float* bv = (const float*)d_in[6];
    const float* Wo = (const float*)d_in[7];
    const float* bo = (const float*)d_in[8];
    float* out = (float*)d_out;

    float* ws   = (float*)d_ws;
    float* Q    = ws;
    float* K    = Q + (size_t)N_ROWS * HEADS;
    float* V    = K + (size_t)N_ROWS * HEADS;
    float* part = V + (size_t)N_ROWS * HEADS;   // SPLITS * N_ROWS * 10 floats

    qkv_kernel<<<dim3(N_ROWS / 256), dim3(256), 0, stream>>>(
        x, Wk, bk, Wq, bq, Wv, bv, Q, K, V);

    flash_kernel<<<dim3(NTILES / WAVES_PER_BLOCK, SPLITS), dim3(256), 0, stream>>>(
        Q, K, V, part);

    combine_kernel<<<dim3(N_ROWS / 256), dim3(256), 0, stream>>>(
        x, Wo, bo, part, out);
}